// GCN_10651518894410
// MI455X (gfx1250) — compile-verified
//
#include <hip/hip_runtime.h>
#include <hip/hip_bf16.h>

typedef __attribute__((ext_vector_type(2))) float v2f;
typedef __attribute__((ext_vector_type(8))) float v8f;

#define N_NODES 100000
#define N_EDGES 1600000
#define IN_F    128
#define HID_F   128
#define OUT_F   64

// ---------------------------------------------------------------- utilities

__global__ void zero_kernel(float* __restrict__ p, long n) {
    long i = (long)blockIdx.x * blockDim.x + threadIdx.x;
    long stride = (long)gridDim.x * blockDim.x;
    for (; i < n; i += stride) p[i] = 0.0f;
}

// in-degree via scatter-add of ones on dst (matches reference segment_sum)
__global__ void deg_count_kernel(const int* __restrict__ dst, float* __restrict__ deg, int E) {
    int e = blockIdx.x * blockDim.x + threadIdx.x;
    if (e < E) atomicAdd(&deg[dst[e]], 1.0f);
}

// deg -> deg^-0.5 in place (0 where deg == 0)
__global__ void deg_finish_kernel(float* __restrict__ deg, int n) {
    int i = blockIdx.x * blockDim.x + threadIdx.x;
    if (i < n) {
        float d = deg[i];
        deg[i] = (d > 0.0f) ? (1.0f / sqrtf(d)) : 0.0f;
    }
}

// ---------------------------------------------------------------- WMMA GEMM
// H[m, n] = (optional relu on X) sum_k X[m,k] * W[n,k] + bias[n]
// X: [N_NODES, 128] row-major, W: [FOUT, 128] row-major (torch Linear layout).
// One wave computes a 16x16 tile; K=128 -> 32 x V_WMMA_F32_16X16X4_F32.
// A/B per-lane layout (ISA 7.12.2, 32-bit 16x4): lanes 0-15 hold K={0,1} in
// VGPR{0,1}, lanes 16-31 hold K={2,3}; lane index = M (for A) / N (for B).
// C/D layout: VGPR r -> M = r + 8*(lane>=16), N = lane%16.
template <int FOUT, bool RELU_IN>
__global__ __launch_bounds__(32 * (FOUT / 16))
void gemm_wmma_kernel(const float* __restrict__ X, const float* __restrict__ W,
                      const float* __restrict__ bias, float* __restrict__ H) {
    constexpr int K = 128;
    const int wave = threadIdx.x >> 5;
    const int lane = threadIdx.x & 31;
    const int half = lane >> 4;   // 0: K base 0, 1: K base 2
    const int l    = lane & 15;   // M index (A) / N index (B)
    const int m0 = blockIdx.x * 16;
    const int n0 = wave * 16;

    const float* arow = X + (long)(m0 + l) * K + 2 * half;
    const float* brow = W + (long)(n0 + l) * K + 2 * half;

    v8f c = {};
#pragma unroll
    for (int s = 0; s < K / 4; ++s) {
        v2f a = *(const v2f*)(arow + 4 * s);
        v2f b = *(const v2f*)(brow + 4 * s);
        if (RELU_IN) {
            a.x = fmaxf(a.x, 0.0f);
            a.y = fmaxf(a.y, 0.0f);
        }
        // (neg_a, A, neg_b, B, c_mod, C, reuse_a, reuse_b)
        c = __builtin_amdgcn_wmma_f32_16x16x4_f32(false, a, false, b,
                                                  (short)0, c, false, false);
    }

    const float bb = bias[n0 + l];
#pragma unroll
    for (int r = 0; r < 8; ++r) {
        const int row = m0 + r + 8 * half;
        H[(long)row * FOUT + n0 + l] = c[r] + bb;
    }
}

// ---------------------------------------------------------------- aggregation
// out[dst] += dis[src]*dis[dst] * h[src]   (scatter-add over edges)
// One wave per edge; 128 feats -> float4 per lane.
__global__ void aggregate128_kernel(const int* __restrict__ src, const int* __restrict__ dst,
                                    const float* __restrict__ dis, const float* __restrict__ h,
                                    float* __restrict__ out, int E) {
    int e = blockIdx.x * (blockDim.x >> 5) + (threadIdx.x >> 5);
    if (e >= E) return;
    int lane = threadIdx.x & 31;
    int s = src[e], d = dst[e];
    float w = dis[s] * dis[d];
    float4 v = ((const float4*)(h + (long)s * 128))[lane];
    float* o = out + (long)d * 128 + lane * 4;
    atomicAdd(o + 0, v.x * w);
    atomicAdd(o + 1, v.y * w);
    atomicAdd(o + 2, v.z * w);
    atomicAdd(o + 3, v.w * w);
}

// 64 feats -> float2 per lane.
__global__ void aggregate64_kernel(const int* __restrict__ src, const int* __restrict__ dst,
                                   const float* __restrict__ dis, const float* __restrict__ h,
                                   float* __restrict__ out, int E) {
    int e = blockIdx.x * (blockDim.x >> 5) + (threadIdx.x >> 5);
    if (e >= E) return;
    int lane = threadIdx.x & 31;
    int s = src[e], d = dst[e];
    float w = dis[s] * dis[d];
    float2 v = ((const float2*)(h + (long)s * 64))[lane];
    float* o = out + (long)d * 64 + lane * 2;
    atomicAdd(o + 0, v.x * w);
    atomicAdd(o + 1, v.y * w);
}

// ---------------------------------------------------------------- launcher

extern "C" void kernel_launch(void* const* d_in, const int* in_sizes, int n_in,
                              void* d_out, int out_size, void* d_ws, size_t ws_size,
                              hipStream_t stream) {
    const float* x  = (const float*)d_in[0];
    const int*   ei = (const int*)d_in[1];   // [2, E]: row0 = src, row1 = dst
    const float* W1 = (const float*)d_in[2];
    const float* b1 = (const float*)d_in[3];
    const float* W2 = (const float*)d_in[4];
    const float* b2 = (const float*)d_in[5];
    float* out = (float*)d_out;

    const int* src = ei;
    const int* dst = ei + N_EDGES;

    // workspace layout (floats): deg/dis [N], h1 [N*128], agg1 [N*128], h2 [N*64]
    float* deg  = (float*)d_ws;
    float* h1   = deg + N_NODES;
    float* agg1 = h1 + (long)N_NODES * HID_F;
    float* h2   = agg1 + (long)N_NODES * HID_F;

    // degree normalization
    zero_kernel<<<1024, 256, 0, stream>>>(deg, (long)N_NODES);
    deg_count_kernel<<<(N_EDGES + 255) / 256, 256, 0, stream>>>(dst, deg, N_EDGES);
    deg_finish_kernel<<<(N_NODES + 255) / 256, 256, 0, stream>>>(deg, N_NODES);

    // layer 1: h1 = x @ W1^T + b1   (100000 = 6250 * 16, no tail)
    gemm_wmma_kernel<HID_F, false><<<N_NODES / 16, 256, 0, stream>>>(x, W1, b1, h1);

    // agg1 = scatter-add of w * h1[src] into dst
    zero_kernel<<<4096, 256, 0, stream>>>(agg1, (long)N_NODES * HID_F);
    aggregate128_kernel<<<(N_EDGES + 7) / 8, 256, 0, stream>>>(src, dst, deg, h1, agg1, N_EDGES);

    // layer 2: h2 = relu(agg1) @ W2^T + b2  (relu fused into A-load)
    gemm_wmma_kernel<OUT_F, true><<<N_NODES / 16, 128, 0, stream>>>(agg1, W2, b2, h2);

    // out = scatter-add of w * h2[src] into dst
    zero_kernel<<<4096, 256, 0, stream>>>(out, (long)N_NODES * OUT_F);
    aggregate64_kernel<<<(N_EDGES + 7) / 8, 256, 0, stream>>>(src, dst, deg, h2, out, N_EDGES);
}